// FractalFlameGenerator_54211077210273
// MI455X (gfx1250) — compile-verified
//
#include <hip/hip_runtime.h>
#include <hip/hip_bf16.h>
#include <stdint.h>
#include <stddef.h>

// ---------------- problem constants (match reference) ----------------
#define BATCH   64
#define KT      7
#define NPARM   10
#define HH      64
#define WW      64
#define PART    2048
#define NITER   64
#define WARMUP  12
#define NOISE   3.0f

#define TPB          512                  // 16 waves (wave32) per block
#define BLKS_PER_IMG (PART / TPB)        // 4 blocks per image -> 256 blocks
#define HIST         (HH * WW * 3)       // 12288 floats = 48 KB in LDS

// ---------------- CDNA5 TDM (tensor DMA) availability ----------------
#if defined(__HIP_DEVICE_COMPILE__) && __has_builtin(__builtin_amdgcn_tensor_load_to_lds)
#define HAVE_TDM 1
typedef __attribute__((ext_vector_type(4))) unsigned int tdm_v4u;
typedef __attribute__((ext_vector_type(8))) int          tdm_v8i;
typedef __attribute__((ext_vector_type(4))) int          tdm_v4i;
#endif

// ---------------- shared memory layout (stage MUST be first: lds_addr 0) ----
struct Smem {
  float stage[80];          // raw u_p row for this image (70 used), TDM target
  float hist[HIST];         // per-block RGB histogram
  float cdf[KT];            // categorical CDF of softmaxed weights
  float prm[KT][NPARM];     // noisy transform params
  float col[KT];            // noisy colours
  float tmp[KT];            // softmax logits scratch
};

// ---------------- stateless counter-based RNG ----------------
__device__ __forceinline__ unsigned hash32(unsigned v) {
  v ^= v >> 16; v *= 0x7FEB352Du;
  v ^= v >> 15; v *= 0x846CA68Bu;
  v ^= v >> 16;
  return v;
}
__device__ __forceinline__ float u01(unsigned s) {
  return (float)(hash32(s) >> 8) * (1.0f / 16777216.0f);  // [0,1)
}

// ---------------- kernels ----------------
__global__ void zero_kernel(float* __restrict__ p, int n) {
  int i = blockIdx.x * blockDim.x + threadIdx.x;
  if (i < n) p[i] = 0.0f;
}

__global__ __launch_bounds__(TPB)
void chaos_kernel(const float* __restrict__ weights,   // (1,K)
                  const float* __restrict__ params,    // (1,K,10)
                  const float* __restrict__ colours,   // (1,K)
                  const float* __restrict__ u_w,       // (B,K)
                  const float* __restrict__ u_p,       // (B,K,10)
                  const float* __restrict__ u_c,       // (B,K)
                  float* __restrict__ gacc)            // (B,H,W,3) accumulator
{
  __shared__ Smem sm;
  const int b   = blockIdx.x / BLKS_PER_IMG;
  const int sub = blockIdx.x % BLKS_PER_IMG;
  const int tid = threadIdx.x;

  // zero the LDS histogram
  for (int i = tid; i < HIST; i += TPB) sm.hist[i] = 0.0f;

  // ---- stage the per-image u_p row (K*NP = 70 f32) into LDS via TDM ----
#ifdef HAVE_TDM
  if (tid < 32) {  // wave 0 only issues the tensor DMA (EXEC ignored by TDM)
    unsigned long long ga =
        (unsigned long long)(uintptr_t)(u_p + (size_t)b * KT * NPARM);
    tdm_v4u g0;
    g0.x = 1u;                                           // count=1 (valid D#)
    g0.y = (unsigned)__builtin_offsetof(Smem, stage);    // lds_addr = 0
    g0.z = (unsigned)(ga & 0xFFFFFFFFu);                 // global_addr[31:0]
    g0.w = (unsigned)((ga >> 32) & 0x01FFFFFFu)          // global_addr[56:32]
         | 0x80000000u;                                  // type = 2 ("image")
    tdm_v8i g1;
    g1[0] = (int)(2u << 16);        // data_size = 4 bytes, no multicast
    g1[1] = (int)(70u << 16);       // tensor_dim0 = 70 (low 16 bits)
    g1[2] = (int)(1u << 16);        // tensor_dim1 = 1
    g1[3] = (int)(70u << 16);       // tile_dim0 = 70
    g1[4] = 1;                      // tile_dim1 = 1, tile_dim2 = 0
    g1[5] = 70;                     // tensor_dim0_stride = 70
    g1[6] = 0;
    g1[7] = 0;
    tdm_v4i z4a = {0, 0, 0, 0};     // group 2 unused (<=2D tensor)
    tdm_v4i z4b = {0, 0, 0, 0};     // group 3 unused
    tdm_v8i z8  = {0, 0, 0, 0, 0, 0, 0, 0};  // extra group (clang-23 form)
    __builtin_amdgcn_tensor_load_to_lds(g0, g1, z4a, z4b, z8, 0);
#if __has_builtin(__builtin_amdgcn_s_wait_tensorcnt)
    __builtin_amdgcn_s_wait_tensorcnt(0);
#else
    asm volatile("s_wait_tensorcnt 0x0" ::: "memory");
#endif
  }
#else
  if (tid < KT * NPARM) sm.stage[tid] = u_p[(size_t)b * KT * NPARM + tid];
#endif

  if (tid < KT) sm.tmp[tid] = weights[tid] * NOISE * u_w[b * KT + tid];
  if (tid < KT)
    sm.col[tid] = fminf(fmaxf(colours[tid] * NOISE * u_c[b * KT + tid], 0.0f), 1.0f);
  __syncthreads();

  // noisy params from staged u_p
  if (tid < KT * NPARM) sm.prm[tid / NPARM][tid % NPARM] =
      params[tid] * NOISE * sm.stage[tid];

  // softmax -> CDF (K=7, single lane)
  if (tid == 0) {
    float m = sm.tmp[0];
    for (int k = 1; k < KT; ++k) m = fmaxf(m, sm.tmp[k]);
    float e[KT], s = 0.0f;
    for (int k = 0; k < KT; ++k) { e[k] = __expf(sm.tmp[k] - m); s += e[k]; }
    float inv = 1.0f / s, run = 0.0f;
    for (int k = 0; k < KT; ++k) { run += e[k] * inv; sm.cdf[k] = run; }
    sm.cdf[KT - 1] = 1.0f;
  }
  __syncthreads();

  // hoist the CDF into registers: the ds_add_f32 atomics below would otherwise
  // force a re-load from LDS every iteration (same-LDS aliasing).
  const float c0 = sm.cdf[0], c1 = sm.cdf[1], c2 = sm.cdf[2];
  const float c3 = sm.cdf[3], c4 = sm.cdf[4], c5 = sm.cdf[5];

  // ---- chaos game: 1 particle per lane ----
  const int p = sub * TPB + tid;                       // particle id in image
  const unsigned base = (unsigned)(b * PART + p);
  float x = u01(base ^ 0xA511E9B3u) * 2.0f - 1.0f;
  float y = u01(base ^ 0x63D83595u) * 2.0f - 1.0f;
  float c = u01(base ^ 0x2545F491u);

  for (int it = 0; it < NITER; ++it) {
    float r = u01(base * 0x9E3779B9u + (unsigned)it * 0x85EBCA77u + 0x165667B1u);
    int k = (r >= c0) + (r >= c1) + (r >= c2) + (r >= c3) + (r >= c4) + (r >= c5);

    const float* pk = sm.prm[k];
    const float a  = pk[0], bb = pk[1], t1 = pk[2];
    const float d  = pk[3], e  = pk[4], t2 = pk[5];
    const float p6 = pk[6], p7 = pk[7], p8 = pk[8], p9 = pk[9];
    const float colk = sm.col[k];

    const float xn = a * x + bb * y + t1;
    const float yn = d * x + e * y + t2;
    const float r2 = xn * xn + yn * yn + 1e-9f;
    const float sr = __sinf(r2), cr = __cosf(r2);
    const float inv = __builtin_amdgcn_rcpf(r2);
    float X = p6 * xn + p7 * __sinf(xn) + p8 * xn * inv + p9 * (xn * sr - yn * cr);
    float Y = p6 * yn + p7 * __sinf(yn) + p8 * yn * inv + p9 * (xn * cr + yn * sr);
    X = fminf(fmaxf(X, -1e4f), 1e4f);
    Y = fminf(fmaxf(Y, -1e4f), 1e4f);
    c = 0.5f * (c + colk);

    if (it >= WARMUP && fabsf(X) < 2.0f && fabsf(Y) < 2.0f) {
      int px = min(max((int)((X + 2.0f) * (WW / 4.0f)), 0), WW - 1);
      int py = min(max((int)((Y + 2.0f) * (HH / 4.0f)), 0), HH - 1);
      int h = (py * WW + px) * 3;
      float g1c = 1.0f - c;
      atomicAdd(&sm.hist[h + 0], c);              // ds_add_f32
      atomicAdd(&sm.hist[h + 1], g1c);
      atomicAdd(&sm.hist[h + 2], 4.0f * c * g1c);
    }
    x = X; y = Y;
  }
  __syncthreads();

  // merge block-private histogram into the per-image global accumulator
  float* dst = gacc + (size_t)b * HIST;
  for (int i = tid; i < HIST; i += TPB) {
    float v = sm.hist[i];
    if (v != 0.0f) atomicAdd(&dst[i], v);         // global_atomic_add_f32
  }
}

__global__ void finalize_kernel(const float* __restrict__ acc,  // (B,H,W,3)
                                float* __restrict__ out)        // (B,3,H,W)
{
  int i = blockIdx.x * blockDim.x + threadIdx.x;
  if (i >= BATCH * 3 * HH * WW) return;
  int xx = i & (WW - 1);
  int yy = (i >> 6) & (HH - 1);
  int ch = (i >> 12) % 3;
  int b  = i / (3 * HH * WW);
  float v = acc[((size_t)(b * HH + yy) * WW + xx) * 3 + ch];
  out[i] = __logf(1.0f + v);                      // log1p, v >= 0
}

// ---------------- harness entry ----------------
extern "C" void kernel_launch(void* const* d_in, const int* in_sizes, int n_in,
                              void* d_out, int out_size, void* d_ws, size_t ws_size,
                              hipStream_t stream) {
  const float* weights = (const float*)d_in[0];
  const float* params  = (const float*)d_in[1];
  const float* colours = (const float*)d_in[2];
  const float* u_w     = (const float*)d_in[3];
  const float* u_p     = (const float*)d_in[4];
  const float* u_c     = (const float*)d_in[5];
  float* out = (float*)d_out;
  float* acc = (float*)d_ws;                       // BATCH*HIST f32 = 3 MB

  const int accN = BATCH * HIST;
  zero_kernel<<<(accN + 255) / 256, 256, 0, stream>>>(acc, accN);
  chaos_kernel<<<BATCH * BLKS_PER_IMG, TPB, 0, stream>>>(
      weights, params, colours, u_w, u_p, u_c, acc);
  const int outN = BATCH * 3 * HH * WW;
  finalize_kernel<<<(outN + 255) / 256, 256, 0, stream>>>(acc, out);
}